// MemStream_63883343561416
// MI455X (gfx1250) — compile-verified
//
#include <hip/hip_runtime.h>
#include <hip/hip_bf16.h>
#include <stdint.h>

// Problem constants (from reference)
constexpr int kInDim  = 256;
constexpr int kCode   = 128;
constexpr int kMemLen = 65536;

typedef float v2f __attribute__((ext_vector_type(2)));
typedef float v8f __attribute__((ext_vector_type(8)));

// Workspace layout (< 8 KB)
struct Ws {
  double sum[kInDim];            // column sums of mem_data
  double sumsq[kInDim];          // column sums of squares
  float  enc[kCode];             // encoded vector
  unsigned int lossBits;         // min L1 dist, float bits (dist >= 0 so uint order == float order)
  unsigned int pad0;
  unsigned long long idxPack;    // (order-mapped mem_idx value << 32) | index -> argmin, first-index tiebreak
};

// ---------------------------------------------------------------- init
__global__ void k_init(Ws* ws) {
  int t = threadIdx.x;
  if (t < kInDim) { ws->sum[t] = 0.0; ws->sumsq[t] = 0.0; }
  if (t == 0) { ws->lossBits = 0x7F7FFFFFu; ws->idxPack = ~0ull; }
}

// ---------------------------------------------- stats over mem_data + copy out
// b128 loads: thread t owns columns c4..c4+3 (c4 = (t&63)*4), rows advance 4 per
// iteration (sub-row t>>6). Per-thread f64 accumulators -> LDS ds_add_f64 combine
// -> one global_atomic_add_f64 pair per column per block.
__global__ void k_stats_copy(const float* __restrict__ mem_data,
                             float* __restrict__ out_mem_data,
                             Ws* __restrict__ ws) {
  __shared__ double sSum[kInDim];
  __shared__ double sSq[kInDim];
  const int t = threadIdx.x;
  sSum[t] = 0.0; sSq[t] = 0.0;
  __syncthreads();

  const int c4   = (t & 63) * 4;        // fixed 4-column group
  const int rsub = t >> 6;              // 0..3
  const int r0   = blockIdx.x * 64;     // 64 rows per block, 1024 blocks

  double s0 = 0.0, s1 = 0.0, s2 = 0.0, s3 = 0.0;
  double q0 = 0.0, q1 = 0.0, q2 = 0.0, q3 = 0.0;

  #pragma unroll 4
  for (int i = 0; i < 16; ++i) {
    const int r = r0 + i * 4 + rsub;
    const size_t off = (size_t)r * kInDim + c4;
    __builtin_prefetch(mem_data + off + 4 * (size_t)kInDim, 0, 1); // next iter, speculative
    const float4 v = *(const float4*)(mem_data + off);
    float* o = out_mem_data + off;      // out base is +1 float -> scalar stores
    o[0] = v.x; o[1] = v.y; o[2] = v.z; o[3] = v.w;
    s0 += (double)v.x; q0 += (double)v.x * (double)v.x;
    s1 += (double)v.y; q1 += (double)v.y * (double)v.y;
    s2 += (double)v.z; q2 += (double)v.z * (double)v.z;
    s3 += (double)v.w; q3 += (double)v.w * (double)v.w;
  }
  atomicAdd(&sSum[c4 + 0], s0); atomicAdd(&sSq[c4 + 0], q0);   // ds_add_f64
  atomicAdd(&sSum[c4 + 1], s1); atomicAdd(&sSq[c4 + 1], q1);
  atomicAdd(&sSum[c4 + 2], s2); atomicAdd(&sSq[c4 + 2], q2);
  atomicAdd(&sSum[c4 + 3], s3); atomicAdd(&sSq[c4 + 3], q3);
  __syncthreads();
  atomicAdd(&ws->sum[t],   sSum[t]);    // global_atomic_add_f64
  atomicAdd(&ws->sumsq[t], sSq[t]);
}

// ---------------------------------------------- normalize + WMMA encode
// 1 block, 8 waves; wave w computes enc[16w .. 16w+15] with V_WMMA_F32_16X16X4_F32.
// A (16x4): row M=0 carries new[k..k+3]; built with an unconditional ds_load_b64
//           times a per-lane 0/1 selector (no EXEC predication in the WMMA loop).
// B (4x16): column n = W row (16w+n).
// D: M=0 row lands in acc[0] on lanes 0..15 (N = lane).
__global__ void k_encode(const float* __restrict__ x,
                         const float* __restrict__ W,
                         const float* __restrict__ bias,
                         Ws* __restrict__ ws) {
  __shared__ float sNew[kInDim];
  const int t = threadIdx.x;
  {
    const double N  = (double)kMemLen;
    const double s  = ws->sum[t];
    const double s2 = ws->sumsq[t];
    const double mean = s / N;
    double var = (s2 - s * s / N) / (N - 1.0);   // ddof = 1
    if (var < 0.0) var = 0.0;
    const float stdv = (float)sqrt(var);
    sNew[t] = (stdv == 0.0f) ? 0.0f : ((x[t] - (float)mean) / stdv);
  }
  __syncthreads();

  const int   lane   = t & 31;
  const int   wave   = t >> 5;             // output tile 0..7
  const int   m      = lane & 15;          // M index (A) == N index (B)
  const int   khalf  = lane >> 4;          // 0 -> K{0,1}, 1 -> K{2,3}
  const int   wrow   = wave * 16 + m;      // W row feeding output column n = m
  const float rowSel = (m == 0) ? 1.0f : 0.0f;

  v8f acc = {};
  for (int k0 = 0; k0 < kInDim; k0 += 4) {
    const int kk = k0 + khalf * 2;
    v2f a, bm;
    a.x  = sNew[kk]     * rowSel;          // unconditional LDS load, VALU select
    a.y  = sNew[kk + 1] * rowSel;
    bm.x = W[(size_t)wrow * kInDim + kk];
    bm.y = W[(size_t)wrow * kInDim + kk + 1];
    acc = __builtin_amdgcn_wmma_f32_16x16x4_f32(false, a, false, bm,
                                                (short)0, acc, false, false);
  }
  if (lane < 16) {
    const int j = wave * 16 + lane;
    ws->enc[j] = acc[0] + bias[j];
  }
}

// ---------------------------------------------- L1 distances over memory + copy out
// one row (128 floats) per wave per iteration; b128 loads, wave32 shfl reduce.
__global__ void k_dist_copy(const float* __restrict__ memory,
                            float* __restrict__ out_memory,
                            Ws* __restrict__ ws) {
  __shared__ float sEnc[kCode];
  __shared__ float sMin[8];
  const int t = threadIdx.x;
  if (t < kCode) sEnc[t] = ws->enc[t];
  __syncthreads();

  const int lane   = t & 31;
  const int wave   = t >> 5;
  const int waveId = blockIdx.x * 8 + wave;
  const int nWaves = gridDim.x * 8;

  const float e0 = sEnc[lane * 4 + 0];
  const float e1 = sEnc[lane * 4 + 1];
  const float e2 = sEnc[lane * 4 + 2];
  const float e3 = sEnc[lane * 4 + 3];

  float best = 3.4e38f;
  for (int row = waveId; row < kMemLen; row += nWaves) {
    const size_t off = (size_t)row * kCode + lane * 4;
    __builtin_prefetch(memory + off + (size_t)nWaves * kCode, 0, 1); // next iter, speculative
    const float4 v = *(const float4*)(memory + off);
    float* o = out_memory + off;          // out base is +1 float -> scalar stores
    o[0] = v.x; o[1] = v.y; o[2] = v.z; o[3] = v.w;
    float d = fabsf(v.x - e0) + fabsf(v.y - e1) + fabsf(v.z - e2) + fabsf(v.w - e3);
    #pragma unroll
    for (int off32 = 16; off32 >= 1; off32 >>= 1)
      d += __shfl_xor(d, off32, 32);
    best = fminf(best, d);
  }
  if (lane == 0) sMin[wave] = best;
  __syncthreads();
  if (t == 0) {
    float m = sMin[0];
    #pragma unroll
    for (int i = 1; i < 8; ++i) m = fminf(m, sMin[i]);
    atomicMin(&ws->lossBits, __float_as_uint(m));   // m >= 0 -> order-preserving
  }
}

// ---------------------------------------------- mem_idx copy + argmin (first-min)
__global__ void k_idx(const int* __restrict__ mem_idx,
                      int* __restrict__ out_idx,
                      Ws* __restrict__ ws) {
  __shared__ unsigned long long sKey[256];
  const int t = threadIdx.x;
  const int i = blockIdx.x * 256 + t;
  const int v = mem_idx[i];
  out_idx[i] = v;
  const unsigned int mv = (unsigned int)v ^ 0x80000000u;        // int32 -> order-preserving uint32
  sKey[t] = ((unsigned long long)mv << 32) | (unsigned int)i;   // ties -> smallest index wins
  __syncthreads();
  #pragma unroll
  for (int off = 128; off >= 1; off >>= 1) {
    if (t < off) {
      const unsigned long long o = sKey[t + off];
      if (o < sKey[t]) sKey[t] = o;
    }
    __syncthreads();
  }
  if (t == 0) atomicMin(&ws->idxPack, sKey[0]);
}

// ---------------------------------------------- finalize: loss + conditional row patch
__global__ void k_finalize(const float* __restrict__ x,
                           const int* __restrict__ count,
                           float* __restrict__ out_loss,
                           float* __restrict__ out_memory,
                           float* __restrict__ out_mem_data,
                           int* __restrict__ out_idx,
                           Ws* __restrict__ ws) {
  const int t = threadIdx.x;
  const float loss = __uint_as_float(ws->lossBits);
  if (t == 0) out_loss[0] = loss;
  if (loss <= 1.0f) {   // BETA
    const unsigned int pos = (unsigned int)(ws->idxPack & 0xFFFFFFFFull);
    if (t < kCode) out_memory[(size_t)pos * kCode + t] = ws->enc[t];
    out_mem_data[(size_t)pos * kInDim + t] = x[t];
    if (t == 0) out_idx[pos] = count[0];
  }
}

// ---------------------------------------------------------------- launch
extern "C" void kernel_launch(void* const* d_in, const int* in_sizes, int n_in,
                              void* d_out, int out_size, void* d_ws, size_t ws_size,
                              hipStream_t stream) {
  const float* x        = (const float*)d_in[0];
  const float* memory   = (const float*)d_in[1];
  const float* mem_data = (const float*)d_in[2];
  const int*   mem_idx  = (const int*)d_in[3];
  const float* W        = (const float*)d_in[4];
  const float* bias     = (const float*)d_in[5];
  const int*   count    = (const int*)d_in[6];

  float* out          = (float*)d_out;
  float* out_loss     = out;
  float* out_memory   = out + 1;
  float* out_mem_data = out + 1 + (size_t)kMemLen * kCode;
  int*   out_idx      = (int*)(out + 1 + (size_t)kMemLen * kCode + (size_t)kMemLen * kInDim);

  Ws* ws = (Ws*)d_ws;

  k_init      <<<1, 256, 0, stream>>>(ws);
  k_stats_copy<<<kMemLen / 64, 256, 0, stream>>>(mem_data, out_mem_data, ws);
  k_encode    <<<1, 256, 0, stream>>>(x, W, bias, ws);
  k_dist_copy <<<512, 256, 0, stream>>>(memory, out_memory, ws);
  k_idx       <<<kMemLen / 256, 256, 0, stream>>>(mem_idx, out_idx, ws);
  k_finalize  <<<1, 256, 0, stream>>>(x, count, out_loss, out_memory, out_mem_data, out_idx, ws);
}